// FourierSparseAttention_20658792693857
// MI455X (gfx1250) — compile-verified
//
#include <hip/hip_runtime.h>
#include <math.h>

#define Bc   4
#define Lc   1024
#define Sc   1024
#define Hc   16
#define Ec   64
#define TOPK 10
#define NFREQ 513            // S/2 + 1
#define EGRP 8               // e-columns per FFT block

typedef float v2f __attribute__((ext_vector_type(2)));
typedef float v8f __attribute__((ext_vector_type(8)));

// ---------------------------------------------------------------------------
// Kernel 1: radix-2 1024-pt FFT per (bh, e); accumulate |F[f]| over 8 e's.
// grid = (8, 64), block = 128.  magse layout: [bh][egrp][513]
// ---------------------------------------------------------------------------
__global__ __launch_bounds__(128)
void fsa_fft_mag(const float* __restrict__ keys, float* __restrict__ magse) {
    __shared__ float re[1024];
    __shared__ float im[1024];
    const int bh = blockIdx.y;
    const int b  = bh >> 4, h = bh & 15;           // H = 16
    const int tid = threadIdx.x;

    float acc[5];                                   // ceil(513/128) = 5 per thread
    #pragma unroll
    for (int q = 0; q < 5; ++q) acc[q] = 0.f;

    for (int ei = 0; ei < EGRP; ++ei) {
        const int e = blockIdx.x * EGRP + ei;
        // load with bit-reversal (10-bit)
        for (int i = tid; i < 1024; i += 128) {
            unsigned rev = __brev((unsigned)i) >> 22;
            re[rev] = keys[(((size_t)b * Sc + i) * Hc + h) * Ec + e];
            im[rev] = 0.f;
        }
        __syncthreads();
        // 10 radix-2 stages
        for (int s = 1; s <= 10; ++s) {
            const int half = 1 << (s - 1);
            for (int k = tid; k < 512; k += 128) {
                const int j    = k & (half - 1);
                const int base = (k >> (s - 1)) << s;
                const int p1 = base + j, p2 = p1 + half;
                const float ang = -(float)M_PI * (float)j / (float)half;
                float sn, cs; sincosf(ang, &sn, &cs);
                const float tr = cs * re[p2] - sn * im[p2];
                const float ti = cs * im[p2] + sn * re[p2];
                const float r1 = re[p1], i1 = im[p1];
                re[p2] = r1 - tr; im[p2] = i1 - ti;
                re[p1] = r1 + tr; im[p1] = i1 + ti;
            }
            __syncthreads();
        }
        // accumulate magnitudes (norm factors don't affect top-k ordering)
        #pragma unroll
        for (int q = 0; q < 5; ++q) {
            const int f = tid + 128 * q;
            if (f < NFREQ) acc[q] += sqrtf(re[f] * re[f] + im[f] * im[f]);
        }
        __syncthreads();                            // re/im reused next ei
    }
    float* dst = magse + ((size_t)bh * EGRP + blockIdx.x) * NFREQ;
    #pragma unroll
    for (int q = 0; q < 5; ++q) {
        const int f = tid + 128 * q;
        if (f < NFREQ) dst[f] = acc[q];
    }
}

// ---------------------------------------------------------------------------
// Kernel 2: deterministic reduce over e-groups + serial top-k (ties -> lowest
// index, matching jax.lax.top_k).  grid = 64, block = 256.
// ---------------------------------------------------------------------------
__global__ __launch_bounds__(256)
void fsa_topk(const float* __restrict__ magse, int* __restrict__ topidx) {
    __shared__ float m[NFREQ];
    const int bh = blockIdx.x, tid = threadIdx.x;
    for (int f = tid; f < NFREQ; f += 256) {
        float s = 0.f;
        const float* src = magse + (size_t)bh * EGRP * NFREQ + f;
        #pragma unroll
        for (int g = 0; g < EGRP; ++g) s += src[(size_t)g * NFREQ];
        m[f] = s;
    }
    __syncthreads();
    if (tid == 0) {
        for (int r = 0; r < TOPK; ++r) {
            int best = 0; float bv = -INFINITY;
            for (int f = 0; f < NFREQ; ++f)
                if (m[f] > bv) { bv = m[f]; best = f; }   // strict > : lowest idx wins
            m[best] = -INFINITY;
            topidx[bh * TOPK + r] = best;
        }
    }
}

// ---------------------------------------------------------------------------
// Kernel 3: zero-fill the dense attn output (256 MB) with float4 stores.
// ---------------------------------------------------------------------------
__global__ __launch_bounds__(256)
void fsa_zero_fill(float4* __restrict__ p, size_t n4) {
    size_t i = (size_t)blockIdx.x * blockDim.x + threadIdx.x;
    const size_t stride = (size_t)gridDim.x * blockDim.x;
    float4 z; z.x = z.y = z.z = z.w = 0.f;
    for (; i < n4; i += stride) p[i] = z;
}

// ---------------------------------------------------------------------------
// Kernel 4: WMMA attention over the 10 selected keys (padded to 16).
// grid = (L/128, B*H), block = 256 (8 waves, 1 wave per 16-row L tile).
// ---------------------------------------------------------------------------
__global__ __launch_bounds__(256)
void fsa_attn_wmma(const float* __restrict__ Q, const float* __restrict__ Kf,
                   const float* __restrict__ Vf, const int* __restrict__ topidx,
                   float* __restrict__ outp, float* __restrict__ attnp) {
    __shared__ float Ks[16][Ec];        // top-k keys   [s'][e], rows >=10 zero
    __shared__ float Vs[16][Ec];        // top-k values [s'][e]
    __shared__ int   idxs[16];
    __shared__ float Qs[8][16][Ec];     // per-wave Q tile [l][e]
    __shared__ float Ps[8][16][17];     // per-wave softmax tile [l][s'] (+pad)

    const int bh = blockIdx.y, b = bh >> 4, h = bh & 15;
    const int wave = threadIdx.x >> 5, lane = threadIdx.x & 31;
    const int lo = lane & 15, hi = lane >> 4;
    const int l0 = blockIdx.x * 128 + wave * 16;

    // gather top-k K/V rows (coalesced over e)
    for (int i = threadIdx.x; i < 16 * Ec; i += 256) {
        const int r = i >> 6, e = i & 63;
        float kv = 0.f, vv = 0.f;
        if (r < TOPK) {
            const int s = topidx[bh * TOPK + r];
            const size_t off = (((size_t)b * Sc + s) * Hc + h) * Ec + e;
            kv = Kf[off]; vv = Vf[off];
        }
        Ks[r][e] = kv; Vs[r][e] = vv;
    }
    if (threadIdx.x < 16)
        idxs[threadIdx.x] = (threadIdx.x < TOPK) ? topidx[bh * TOPK + threadIdx.x] : 0;
    // each wave stages its 16 Q rows (coalesced over e)
    for (int i = lane; i < 16 * Ec; i += 32) {
        const int r = i >> 6, e = i & 63;
        Qs[wave][r][e] = Q[(((size_t)b * Lc + (l0 + r)) * Hc + h) * Ec + e];
    }
    __syncthreads();

    // ---- scores = Q @ K_topk^T : 16 x v_wmma_f32_16x16x4_f32 over K=64 ----
    v8f c = {0.f, 0.f, 0.f, 0.f, 0.f, 0.f, 0.f, 0.f};
    #pragma unroll
    for (int kc = 0; kc < 16; ++kc) {
        const int k0 = 4 * kc + 2 * hi;
        v2f a;  a.x  = Qs[wave][lo][k0]; a.y  = Qs[wave][lo][k0 + 1];   // A[M=lo][K]
        v2f bm; bm.x = Ks[lo][k0];       bm.y = Ks[lo][k0 + 1];         // B[K][N=lo]
        c = __builtin_amdgcn_wmma_f32_16x16x4_f32(false, a, false, bm,
                                                  (short)0, c, false, false);
    }

    // ---- scale, mask cols >= TOPK, row-softmax across 16-lane half ----
    const float scale = 0.125f;          // 1/sqrt(64)
    float p[8];
    #pragma unroll
    for (int v = 0; v < 8; ++v) {
        float sc = c[v] * scale;
        if (lo >= TOPK) sc = -INFINITY;
        float mx = sc;
        #pragma unroll
        for (int msk = 1; msk <= 8; msk <<= 1) mx = fmaxf(mx, __shfl_xor(mx, msk, 32));
        const float ex = expf(sc - mx);  // exp(-inf) = 0 on masked cols
        float sm = ex;
        #pragma unroll
        for (int msk = 1; msk <= 8; msk <<= 1) sm += __shfl_xor(sm, msk, 32);
        p[v] = ex / sm;
    }

    // ---- scatter nonzero attn entries + stage P for the A-side of P@V ----
    #pragma unroll
    for (int v = 0; v < 8; ++v) {
        const int row = v + 8 * hi;                       // C layout: row = v + 8*hi
        Ps[wave][row][lo] = p[v];
        if (lo < TOPK)
            attnp[((size_t)bh * Lc + (l0 + row)) * Sc + idxs[lo]] = p[v];
    }
    __syncthreads();

    // ---- out = P @ V_topk : 4 N-tiles x 4 K-steps of WMMA ----
    #pragma unroll
    for (int nt = 0; nt < 4; ++nt) {
        v8f o = {0.f, 0.f, 0.f, 0.f, 0.f, 0.f, 0.f, 0.f};
        #pragma unroll
        for (int kc = 0; kc < 4; ++kc) {
            const int k0 = 4 * kc + 2 * hi;
            v2f a;  a.x  = Ps[wave][lo][k0];      a.y  = Ps[wave][lo][k0 + 1];
            v2f bm; bm.x = Vs[k0][nt * 16 + lo];  bm.y = Vs[k0 + 1][nt * 16 + lo];
            o = __builtin_amdgcn_wmma_f32_16x16x4_f32(false, a, false, bm,
                                                      (short)0, o, false, false);
        }
        #pragma unroll
        for (int v = 0; v < 8; ++v) {
            const int row = v + 8 * hi;
            outp[(((size_t)b * Lc + (l0 + row)) * Hc + h) * Ec + nt * 16 + lo] = o[v];
        }
    }
}

// ---------------------------------------------------------------------------
extern "C" void kernel_launch(void* const* d_in, const int* in_sizes, int n_in,
                              void* d_out, int out_size, void* d_ws, size_t ws_size,
                              hipStream_t stream) {
    const float* Q  = (const float*)d_in[0];
    const float* Kf = (const float*)d_in[1];
    const float* Vf = (const float*)d_in[2];

    float* outp  = (float*)d_out;                                 // [B,L,H,E]
    float* attnp = outp + (size_t)Bc * Lc * Hc * Ec;              // [B,H,L,S]

    float* magse  = (float*)d_ws;                                 // [64][8][513]
    int*   topidx = (int*)((char*)d_ws +
                           (size_t)Bc * Hc * EGRP * NFREQ * sizeof(float));

    fsa_fft_mag<<<dim3(Ec / EGRP, Bc * Hc), 128, 0, stream>>>(Kf, magse);
    fsa_topk<<<Bc * Hc, 256, 0, stream>>>(magse, topidx);

    const size_t n4 = (size_t)Bc * Hc * Lc * Sc / 4;
    fsa_zero_fill<<<4096, 256, 0, stream>>>((float4*)attnp, n4);

    fsa_attn_wmma<<<dim3(Lc / 128, Bc * Hc), 256, 0, stream>>>(
        Q, Kf, Vf, topidx, outp, attnp);
}